// GraphAttentionLayer_9998683865215
// MI455X (gfx1250) — compile-verified
//
#include <hip/hip_runtime.h>
#include <hip/hip_bf16.h>
#include <math.h>

#define NN 1024
#define NODE_DIM 128
#define HIDDEN 128
#define HEADS 4
#define HD 32
#define EDGE_DIM 16

typedef __attribute__((ext_vector_type(2))) float v2f;
typedef __attribute__((ext_vector_type(8))) float v8f;

static __device__ __forceinline__ v8f wmma4(v2f a, v2f b, v8f c) {
  // V_WMMA_F32_16X16X4_F32 : D(16x16 f32) = A(16x4 f32) * B(4x16 f32) + C
  return __builtin_amdgcn_wmma_f32_16x16x4_f32(false, a, false, b, (short)0, c,
                                               false, false);
}

// ---------------------------------------------------------------------------
// Kernel 1: Q/K/V projections. grid = (64 i-tiles, 3 matrices), block = 256.
// Each wave computes one 16x16 output tile; K loop = 128 in steps of 4.
// ---------------------------------------------------------------------------
__global__ void k_proj(const float* __restrict__ nodes,
                       const float* __restrict__ Wq,
                       const float* __restrict__ Wk,
                       const float* __restrict__ Wv,
                       float* __restrict__ qkv_base) {
  const int it = blockIdx.x;
  const int mat = blockIdx.y;  // 0=Q 1=K 2=V
  const float* W = (mat == 0) ? Wq : (mat == 1) ? Wk : Wv;
  float* Out = qkv_base + (size_t)mat * NN * HIDDEN;
  const int wave = threadIdx.x >> 5;
  const int lane = threadIdx.x & 31;
  const int half = lane >> 4, lr = lane & 15, klo = half * 2;
  const int i0 = it * 16;
  const int n0 = wave * 16;
  v8f c = {};
  for (int k0 = 0; k0 < NODE_DIM; k0 += 4) {
    v2f a, b;
    const float* ap = nodes + (size_t)(i0 + lr) * NODE_DIM + k0 + klo;
    a.x = ap[0];
    a.y = ap[1];
    const float* bp = W + (size_t)(k0 + klo) * HIDDEN + n0 + lr;
    b.x = bp[0];
    b.y = bp[HIDDEN];
    c = wmma4(a, b, c);
  }
#pragma unroll
  for (int v = 0; v < 8; ++v)
    Out[(size_t)(i0 + v + half * 8) * HIDDEN + n0 + lr] = c[v];
}

// ---------------------------------------------------------------------------
// Kernel 2: QWe[i][h][c] = sum_d Q[i,h*32+d] * W_e[c,h*32+d]. grid=1024, blk=64.
// ---------------------------------------------------------------------------
__global__ void k_qwe(const float* __restrict__ Q, const float* __restrict__ We,
                      float* __restrict__ QWe) {
  __shared__ float qrow[HIDDEN];
  const int i = blockIdx.x;
  const int t = threadIdx.x;  // 64 threads
  qrow[t] = Q[(size_t)i * HIDDEN + t];
  qrow[t + 64] = Q[(size_t)i * HIDDEN + t + 64];
  __syncthreads();
  const int h = t >> 4, cc = t & 15;
  float s = 0.f;
#pragma unroll
  for (int d = 0; d < HD; ++d)
    s += qrow[h * HD + d] * We[cc * HIDDEN + h * HD + d];
  QWe[(size_t)i * 64 + h * 16 + cc] = s;
}

// ---------------------------------------------------------------------------
// Kernel 3: masked scores. grid = (64 j-tiles, 64 i-tiles), block = 256.
// Waves 0..3: per-head QK^T 16x16 tile via WMMA -> LDS. Then each thread owns
// one (i,j) pair: streams edges[i,j,0:16], dots with QWe, masks, writes score.
// ---------------------------------------------------------------------------
__global__ void k_scores(const float* __restrict__ nodes,
                         const float* __restrict__ edges,
                         const int* __restrict__ adj,
                         const float* __restrict__ Q,
                         const float* __restrict__ K,
                         const float* __restrict__ QWe,
                         float* __restrict__ scores) {
  __shared__ float s_qk[HEADS][16][16];
  __shared__ float s_qwe[16 * 64];
  __shared__ float s_pos[16][4];
  const int jt = blockIdx.x, it = blockIdx.y;
  const int i0 = it * 16, j0 = jt * 16;
  const int t = threadIdx.x;
  const int wave = t >> 5, lane = t & 31;
  const int half = lane >> 4, lr = lane & 15, klo = half * 2;

  for (int k = t; k < 16 * 64; k += 256) s_qwe[k] = QWe[(size_t)i0 * 64 + k];
  if (t < 16) {
    s_pos[t][0] = nodes[(size_t)(i0 + t) * NODE_DIM + 0];
    s_pos[t][1] = nodes[(size_t)(i0 + t) * NODE_DIM + 1];
    s_pos[t][2] = nodes[(size_t)(i0 + t) * NODE_DIM + 2];
  }
  if (wave < HEADS) {
    const int h = wave;
    v8f c = {};
#pragma unroll
    for (int k0 = 0; k0 < HD; k0 += 4) {
      v2f a, b;
      const float* ap = Q + (size_t)(i0 + lr) * HIDDEN + h * HD + k0 + klo;
      a.x = ap[0];
      a.y = ap[1];
      // B[k][n] = K[j0+n][h*32+k]
      const float* bp = K + (size_t)(j0 + lr) * HIDDEN + h * HD + k0 + klo;
      b.x = bp[0];
      b.y = bp[1];
      c = wmma4(a, b, c);
    }
#pragma unroll
    for (int v = 0; v < 8; ++v) s_qk[h][v + half * 8][lr] = c[v];
  }
  __syncthreads();

  const int ti = t >> 4, tj = t & 15;
  const int i = i0 + ti, j = j0 + tj;
  const float4* ep =
      (const float4*)(edges + (size_t)(i * NN + j) * EDGE_DIM);
  float4 e0 = ep[0], e1 = ep[1], e2 = ep[2], e3 = ep[3];
  float eb[16] = {e0.x, e0.y, e0.z, e0.w, e1.x, e1.y, e1.z, e1.w,
                  e2.x, e2.y, e2.z, e2.w, e3.x, e3.y, e3.z, e3.w};
  const float px = nodes[(size_t)j * NODE_DIM + 0] - s_pos[ti][0];
  const float py = nodes[(size_t)j * NODE_DIM + 1] - s_pos[ti][1];
  const float pz = nodes[(size_t)j * NODE_DIM + 2] - s_pos[ti][2];
  const float d2 = px * px + py * py + pz * pz;
  const bool masked =
      (d2 >= 4.0f) || (adj[(size_t)i * NN + j] == 0) || (i == j);
  const float inv_sqrt_d = 0.17677669529663687f;  // 1/sqrt(32)
#pragma unroll
  for (int h = 0; h < HEADS; ++h) {
    float ed = 0.f;
#pragma unroll
    for (int cc = 0; cc < 16; ++cc) ed += eb[cc] * s_qwe[ti * 64 + h * 16 + cc];
    float sc = masked ? -__builtin_inff()
                      : (s_qk[h][ti][tj] + ed) * inv_sqrt_d;
    scores[((size_t)i * HEADS + h) * NN + j] = sc;
  }
}

// ---------------------------------------------------------------------------
// Kernel 4: softmax over j per (i,h); attn in-place; avg_attn to output.
// grid = 1024 (one per i), block = 128 (wave per head).
// ---------------------------------------------------------------------------
__global__ void k_softmax(float* __restrict__ sc, float* __restrict__ avg_out) {
  const int i = blockIdx.x;
  const int t = threadIdx.x;
  const int h = t >> 5, lane = t & 31;
  float* row = sc + ((size_t)i * HEADS + h) * NN;
  float m = -__builtin_inff();
  for (int j = lane; j < NN; j += 32) m = fmaxf(m, row[j]);
  for (int o = 16; o > 0; o >>= 1) m = fmaxf(m, __shfl_xor(m, o, 32));
  float s = 0.f;
  const bool any = (m > -__builtin_inff());
  if (any) {
    for (int j = lane; j < NN; j += 32) s += expf(row[j] - m);
    for (int o = 16; o > 0; o >>= 1) s += __shfl_xor(s, o, 32);
  }
  const float r = (any && s > 0.f) ? 1.0f / s : 0.f;
  for (int j = lane; j < NN; j += 32) {
    float p = any ? expf(row[j] - m) * r : 0.f;
    row[j] = p;
  }
  __syncthreads();
  const float* base = sc + (size_t)i * HEADS * NN;
  for (int j = t; j < NN; j += 128) {
    float a = base[j] + base[NN + j] + base[2 * NN + j] + base[3 * NN + j];
    avg_out[(size_t)i * NN + j] = 0.25f * a;
  }
}

// ---------------------------------------------------------------------------
// Kernel 5: attended = attn @ V (WMMA, K=1024), + residual, LayerNorm.
// grid = 64 (i-tiles), block = 256. Wave w -> head w>>1, 16-col half w&1.
// ---------------------------------------------------------------------------
__global__ void k_out(const float* __restrict__ attn,
                      const float* __restrict__ V,
                      const float* __restrict__ nodes,
                      const float* __restrict__ ln_scale,
                      const float* __restrict__ ln_bias,
                      float* __restrict__ out) {
  __shared__ float x[16][HIDDEN];
  __shared__ float s_mu[16], s_rv[16];
  const int i0 = blockIdx.x * 16;
  const int t = threadIdx.x;
  const int wave = t >> 5, lane = t & 31;
  const int half = lane >> 4, lr = lane & 15, klo = half * 2;
  const int h = wave >> 1, nh = wave & 1;
  const int c0 = h * HD + nh * 16;
  v8f c = {};
  for (int j0 = 0; j0 < NN; j0 += 4) {
    v2f a, b;
    const float* ap = attn + ((size_t)(i0 + lr) * HEADS + h) * NN + j0 + klo;
    a.x = ap[0];
    a.y = ap[1];
    const float* bp = V + (size_t)(j0 + klo) * HIDDEN + c0 + lr;
    b.x = bp[0];
    b.y = bp[HIDDEN];
    c = wmma4(a, b, c);
  }
#pragma unroll
  for (int v = 0; v < 8; ++v) {
    int row = v + half * 8;
    x[row][c0 + lr] = c[v] + nodes[(size_t)(i0 + row) * HIDDEN + c0 + lr];
  }
  __syncthreads();
  if (t < 16) {
    float mu = 0.f;
    for (int cc = 0; cc < HIDDEN; ++cc) mu += x[t][cc];
    mu *= (1.0f / HIDDEN);
    float var = 0.f;
    for (int cc = 0; cc < HIDDEN; ++cc) {
      float d = x[t][cc] - mu;
      var += d * d;
    }
    var *= (1.0f / HIDDEN);
    s_mu[t] = mu;
    s_rv[t] = rsqrtf(var + 1e-6f);
  }
  __syncthreads();
  for (int e = t; e < 16 * HIDDEN; e += 256) {
    int row = e >> 7, cc = e & 127;
    out[(size_t)(i0 + row) * HIDDEN + cc] =
        (x[row][cc] - s_mu[row]) * s_rv[row] * ln_scale[cc] + ln_bias[cc];
  }
}

// ---------------------------------------------------------------------------
extern "C" void kernel_launch(void* const* d_in, const int* in_sizes, int n_in,
                              void* d_out, int out_size, void* d_ws,
                              size_t ws_size, hipStream_t stream) {
  const float* nodes = (const float*)d_in[0];
  const float* edges = (const float*)d_in[1];
  const int* adj = (const int*)d_in[2];
  const float* Wq = (const float*)d_in[3];
  const float* Wk = (const float*)d_in[4];
  const float* Wv = (const float*)d_in[5];
  const float* We = (const float*)d_in[6];
  const float* lns = (const float*)d_in[7];
  const float* lnb = (const float*)d_in[8];
  float* out = (float*)d_out;  // [0 .. 131071] = out, then avg_attn (1024x1024)
  float* ws = (float*)d_ws;

  float* sc = ws;                                   // N*HEADS*N   = 16 MB
  float* Q = ws + (size_t)NN * HEADS * NN;          // N*HIDDEN
  float* K = Q + (size_t)NN * HIDDEN;               // N*HIDDEN
  float* V = K + (size_t)NN * HIDDEN;               // N*HIDDEN
  float* QWe = V + (size_t)NN * HIDDEN;             // N*64

  k_proj<<<dim3(64, 3), 256, 0, stream>>>(nodes, Wq, Wk, Wv, Q);
  k_qwe<<<NN, 64, 0, stream>>>(Q, We, QWe);
  k_scores<<<dim3(64, 64), 256, 0, stream>>>(nodes, edges, adj, Q, K, QWe, sc);
  k_softmax<<<NN, 128, 0, stream>>>(sc, out + (size_t)NN * HIDDEN);
  k_out<<<64, 256, 0, stream>>>(sc, V, nodes, lns, lnb, out);
}